// Net_22445499089309
// MI455X (gfx1250) — compile-verified
//
#include <hip/hip_runtime.h>

// ---------------------------------------------------------------------------
// Word-mean-pool + linear head, fused & reordered for MI455X (gfx1250).
//
// Math: mean-pool is linear, so  head(meanpool(x)) == meanpool(head(x)).
// We project each token to C=3 logits first (WMMA f32 GEMM, head padded
// 3 -> 16 columns), then mean-pool 3 scalars per token instead of 1024.
//
// Roofline: 268 MB streaming read of x dominates (~11.5 us @ 23.3 TB/s);
// 0.4 GFLOP of f32 FMAs ride along via V_WMMA_F32_16X16X4_F32.
// ---------------------------------------------------------------------------

typedef __attribute__((ext_vector_type(2))) float v2f;
typedef __attribute__((ext_vector_type(8))) float v8f;

#define SEQ   2048
#define HID   1024
#define NWORD 512
#define NCLS  3
#define BROWS 4     // head rows staged in LDS: 3 real + 1 zero row for padding

// ---------------------------------------------------------------------------
// Kernel 1: proj[t][c] = sum_h x[t][h] * head_w[c][h]
// One wave (32 lanes) per 16-token tile, K swept 1024 in steps of 4 with
// V_WMMA_F32_16X16X4_F32.  head_w staged in LDS (16 KB: 3 rows + zero row,
// so padded columns read zeros directly -- no per-iteration mask VALU).
//
// A-matrix (16x4 f32) VGPR layout per ISA 7.12.2:
//   lanes 0-15: M=lane, VGPR0=K0, VGPR1=K1 ; lanes 16-31: M=lane-16, K2/K3.
// B-matrix (4x16) mirrored: lane holds column N=lane%16, rows (k+koff, +1).
// C/D (16x16 f32): VGPR r -> M = r (lanes 0-15) / r+8 (lanes 16-31), N=lane%16.
// ---------------------------------------------------------------------------
__global__ __launch_bounds__(256) void proj_wmma_kernel(
    const float* __restrict__ x,        // [ntiles*16, HID]
    const float* __restrict__ head_w,   // [NCLS, HID]
    float* __restrict__ proj,           // [ntiles*16, NCLS]
    int ntiles)
{
    __shared__ float hw_lds[BROWS * HID];           // 16 KB, row-major [c][k]

    const int tid = threadIdx.x;
    for (int i = tid; i < BROWS * HID; i += 256)
        hw_lds[i] = (i < NCLS * HID) ? head_w[i] : 0.0f;
    __syncthreads();

    const int wave = tid >> 5;
    const int lane = tid & 31;
    const int tile = blockIdx.x * 8 + wave;
    if (tile >= ntiles) return;                     // uniform per wave

    const int row  = lane & 15;                     // M index (token in tile)
    const int koff = (lane >> 4) << 1;              // 0 for lanes 0-15, 2 else
    const int n    = lane & 15;                     // output column (class)
    const int nc   = (n < NCLS) ? n : NCLS;         // padded cols -> zero row

    const float* __restrict__ xa =
        x + (size_t)(tile * 16 + row) * HID + koff;
    const float* __restrict__ bb = &hw_lds[nc * HID + koff];

    v8f acc = {};
#pragma unroll 8
    for (int step = 0; step < HID / 4; ++step) {
        v2f a = *(const v2f*)(xa + step * 4);       // global_load_b64, clause-able
        v2f b = *(const v2f*)(bb + step * 4);       // ds_load (zero row pads N>=3)
        // D = A x B + C ; emits v_wmma_f32_16x16x4_f32
        acc = __builtin_amdgcn_wmma_f32_16x16x4_f32(
            /*neg_a=*/false, a, /*neg_b=*/false, b,
            /*c_mod=*/(short)0, acc, /*reuse_a=*/false, /*reuse_b=*/false);
    }

    // Only columns N < 3 are real logit channels.
    if (n < NCLS) {
        const int mbase = tile * 16 + ((lane >> 4) << 3);   // +8 for upper half
#pragma unroll
        for (int r = 0; r < 8; ++r)
            proj[(size_t)(mbase + r) * NCLS + n] = acc[r];
    }
}

// ---------------------------------------------------------------------------
// Kernel 2: per batch row, segment mean over word spans of the 3-wide proj,
// then scatter back + bias.  LDS accumulators + ds_add_f32 atomics.
// ---------------------------------------------------------------------------
__global__ __launch_bounds__(256) void pool_scatter_kernel(
    const float* __restrict__ proj,     // [B*SEQ, NCLS]
    const int*   __restrict__ wid,      // [B, SEQ]
    const float* __restrict__ head_b,   // [NCLS]
    float* __restrict__ out)            // [B, SEQ, NCLS]
{
    __shared__ float sums[NWORD * NCLS];            // 6 KB
    __shared__ float cnts[NWORD];                   // 2 KB

    const int b   = blockIdx.x;
    const int tid = threadIdx.x;

    for (int i = tid; i < NWORD * NCLS; i += 256) sums[i] = 0.0f;
    for (int i = tid; i < NWORD;        i += 256) cnts[i] = 0.0f;
    __syncthreads();

    const int*   __restrict__ widr  = wid  + (size_t)b * SEQ;
    const float* __restrict__ projr = proj + (size_t)b * SEQ * NCLS;

    for (int s = tid; s < SEQ; s += 256) {
        const int w = widr[s];
        atomicAdd(&cnts[w], 1.0f);                  // ds_add_f32
        atomicAdd(&sums[w * NCLS + 0], projr[s * NCLS + 0]);
        atomicAdd(&sums[w * NCLS + 1], projr[s * NCLS + 1]);
        atomicAdd(&sums[w * NCLS + 2], projr[s * NCLS + 2]);
    }
    __syncthreads();

    const float b0 = head_b[0], b1 = head_b[1], b2 = head_b[2];
    float* __restrict__ outr = out + (size_t)b * SEQ * NCLS;

    for (int s = tid; s < SEQ; s += 256) {
        const int w = widr[s];
        const float inv = 1.0f / fmaxf(cnts[w], 1.0f);
        outr[s * NCLS + 0] = sums[w * NCLS + 0] * inv + b0;
        outr[s * NCLS + 1] = sums[w * NCLS + 1] * inv + b1;
        outr[s * NCLS + 2] = sums[w * NCLS + 2] * inv + b2;
    }
}

// ---------------------------------------------------------------------------
// Inputs (setup_inputs order):
//   d_in[0] x        f32 [B,S,H]     d_in[1] word_ids i32 [B,S]
//   d_in[2] head_w   f32 [C,H]       d_in[3] head_b   f32 [C]
// Output: f32 [B,S,C].  Scratch: proj = B*S*C*4 = 768 KB of d_ws, fully
// rewritten every launch (no state carried across calls).
// ---------------------------------------------------------------------------
extern "C" void kernel_launch(void* const* d_in, const int* in_sizes, int n_in,
                              void* d_out, int out_size, void* d_ws, size_t ws_size,
                              hipStream_t stream)
{
    const float* x      = (const float*)d_in[0];
    const int*   wid    = (const int*)  d_in[1];
    const float* head_w = (const float*)d_in[2];
    const float* head_b = (const float*)d_in[3];
    float*       out    = (float*)d_out;
    float*       proj   = (float*)d_ws;            // [B*S, 3]

    const int B      = in_sizes[1] / SEQ;          // 32
    const int ntiles = (B * SEQ) / 16;             // 4096 tiles -> 4096 waves

    proj_wmma_kernel<<<dim3((ntiles + 7) / 8), dim3(256), 0, stream>>>(
        x, head_w, proj, ntiles);

    pool_scatter_kernel<<<dim3(B), dim3(256), 0, stream>>>(
        proj, wid, head_b, out);
}